// HyperedgeGenerator_59012850647225
// MI455X (gfx1250) — compile-verified
//
#include <hip/hip_runtime.h>
#include <math.h>

typedef __attribute__((ext_vector_type(2))) float v2f;
typedef __attribute__((ext_vector_type(8))) float v8f;

#define N_NODES 1024
#define D_DIM   256

// ---------------------------------------------------------------------------
// Kernel 1: f32 WMMA GEMM (V_WMMA_F32_16X16X4_F32), one 16x16 tile per wave.
//   part 0: A[i,p]  = sum_k nodes[i,k] * W1[p, k]
//   part 1: Bb[j,p] = sum_k nodes[j,k] * W1[p, 256+k] + b1[p]
// Fragment layout (ISA 7.12.2, f32): lane L, vgpr v -> K = 2*(L>>4)+v,
// row/col = L&15 (identical pattern for A and B in an NT-GEMM).
// D layout: vgpr r, lane L -> m = r + 8*(L>>4), n = L&15.
// ---------------------------------------------------------------------------
__global__ __launch_bounds__(256) void gemm_ab_wmma(
    const float* __restrict__ nodes, const float* __restrict__ W1,
    const float* __restrict__ b1, float* __restrict__ Aout,
    float* __restrict__ Bout)
{
  const int lane = threadIdx.x & 31;
  const int waveInBlock = threadIdx.x >> 5;
  const int tile = blockIdx.x * 8 + waveInBlock;  // 0..2047 (uniform per wave)
  const int part = tile >> 10;                    // 0 -> A, 1 -> Bb
  const int t    = tile & 1023;
  const int i0   = (t >> 4) << 4;                 // 64 row tiles
  const int p0   = (t & 15) << 4;                 // 16 col tiles

  const int half = lane >> 4;                     // 0 or 1
  const int mrow = lane & 15;

  const float* aPtr = nodes + (i0 + mrow) * D_DIM + 2 * half;
  const float* bPtr = W1 + (p0 + mrow) * (2 * D_DIM) + part * D_DIM + 2 * half;

  v8f acc = {};
#pragma unroll 4
  for (int k = 0; k < D_DIM; k += 4) {
    v2f afrag = *(const v2f*)(aPtr + k);
    v2f bfrag = *(const v2f*)(bPtr + k);
    // 8 args: (neg_a, A, neg_b, B, c_mod, C, reuse_a, reuse_b)
    acc = __builtin_amdgcn_wmma_f32_16x16x4_f32(false, afrag, false, bfrag,
                                                (short)0, acc, false, false);
  }

  float* outP = (part == 0) ? Aout : Bout;
  const int n = mrow;
  const float bias = (part == 1) ? b1[p0 + n] : 0.0f;
#pragma unroll
  for (int r = 0; r < 8; ++r) {
    const int m = r + half * 8;
    outP[(i0 + m) * D_DIM + (p0 + n)] = acc[r] + bias;
  }
}

// ---------------------------------------------------------------------------
// Kernel 2: pairwise ReLU-dot + sigmoid. The ReLU sits between the pairwise
// add and the dot reduction, so this stage cannot be WMMA — it's pure VALU.
// 2x2 register blocking: 256 threads compute a 32x32 (i,j) tile; each d-step
// does 5 ds_load_b128 for 48 VALU ops, with 4 independent accumulator chains.
// Rows padded to 260 floats (1040 B, 16 B aligned) -> conflict-free b128 LDS.
// ---------------------------------------------------------------------------
__global__ __launch_bounds__(256) void pairwise_kernel(
    const float* __restrict__ A, const float* __restrict__ Bb,
    const float* __restrict__ W2, const float* __restrict__ b2,
    float* __restrict__ out)
{
  __shared__ float sa[32][260];
  __shared__ float sb[32][260];
  __shared__ float sw[256];

  const int tx  = threadIdx.x;           // j within tile (0..15)
  const int ty  = threadIdx.y;           // i within tile (0..15)
  const int tid = ty * 16 + tx;
  const int i0  = blockIdx.y * 32;
  const int j0  = blockIdx.x * 32;

  // cooperative loads: 32 rows x 256 cols per tile, 32 floats per thread
  {
    const int row = tid >> 3;            // 0..31
    const int c0  = (tid & 7) * 32;      // 0..224
    const float4* ga = (const float4*)(A  + (i0 + row) * D_DIM + c0);
    const float4* gb = (const float4*)(Bb + (j0 + row) * D_DIM + c0);
    float4* la = (float4*)&sa[row][c0];
    float4* lb = (float4*)&sb[row][c0];
#pragma unroll
    for (int q = 0; q < 8; ++q) { la[q] = ga[q]; lb[q] = gb[q]; }
    if (tid < 64) ((float4*)sw)[tid] = ((const float4*)W2)[tid];
  }
  const float bias2 = b2[0];
  __syncthreads();

  float acc00 = 0.0f, acc01 = 0.0f, acc10 = 0.0f, acc11 = 0.0f;
#pragma unroll 2
  for (int d = 0; d < D_DIM; d += 4) {
    const float4 av0 = *(const float4*)&sa[ty][d];
    const float4 av1 = *(const float4*)&sa[ty + 16][d];
    const float4 bv0 = *(const float4*)&sb[tx][d];
    const float4 bv1 = *(const float4*)&sb[tx + 16][d];
    const float4 wv  = *(const float4*)&sw[d];

    acc00 += fmaxf(av0.x + bv0.x, 0.0f) * wv.x;
    acc01 += fmaxf(av0.x + bv1.x, 0.0f) * wv.x;
    acc10 += fmaxf(av1.x + bv0.x, 0.0f) * wv.x;
    acc11 += fmaxf(av1.x + bv1.x, 0.0f) * wv.x;

    acc00 += fmaxf(av0.y + bv0.y, 0.0f) * wv.y;
    acc01 += fmaxf(av0.y + bv1.y, 0.0f) * wv.y;
    acc10 += fmaxf(av1.y + bv0.y, 0.0f) * wv.y;
    acc11 += fmaxf(av1.y + bv1.y, 0.0f) * wv.y;

    acc00 += fmaxf(av0.z + bv0.z, 0.0f) * wv.z;
    acc01 += fmaxf(av0.z + bv1.z, 0.0f) * wv.z;
    acc10 += fmaxf(av1.z + bv0.z, 0.0f) * wv.z;
    acc11 += fmaxf(av1.z + bv1.z, 0.0f) * wv.z;

    acc00 += fmaxf(av0.w + bv0.w, 0.0f) * wv.w;
    acc01 += fmaxf(av0.w + bv1.w, 0.0f) * wv.w;
    acc10 += fmaxf(av1.w + bv0.w, 0.0f) * wv.w;
    acc11 += fmaxf(av1.w + bv1.w, 0.0f) * wv.w;
  }

  const int i  = i0 + ty;
  const int j  = j0 + tx;
  out[i * N_NODES + j]                  = 1.0f / (1.0f + __expf(-(acc00 + bias2)));
  out[i * N_NODES + j + 16]             = 1.0f / (1.0f + __expf(-(acc01 + bias2)));
  out[(i + 16) * N_NODES + j]           = 1.0f / (1.0f + __expf(-(acc10 + bias2)));
  out[(i + 16) * N_NODES + j + 16]      = 1.0f / (1.0f + __expf(-(acc11 + bias2)));
}

// ---------------------------------------------------------------------------
extern "C" void kernel_launch(void* const* d_in, const int* in_sizes, int n_in,
                              void* d_out, int out_size, void* d_ws,
                              size_t ws_size, hipStream_t stream)
{
  const float* nodes = (const float*)d_in[0];  // (1024, 256)
  const float* W1    = (const float*)d_in[1];  // (256, 512)
  const float* b1    = (const float*)d_in[2];  // (256,)
  const float* W2    = (const float*)d_in[3];  // (1, 256)
  const float* b2    = (const float*)d_in[4];  // (1,)
  float* out = (float*)d_out;                  // (1024, 1024)

  float* wsA = (float*)d_ws;                   // 1024*256 f32
  float* wsB = wsA + N_NODES * D_DIM;          // 1024*256 f32 (b + b1 fused)

  // 2048 tiles total (2 matrices x 64 x 16), 8 waves per 256-thread block.
  gemm_ab_wmma<<<256, 256, 0, stream>>>(nodes, W1, b1, wsA, wsB);

  dim3 grid(32, 32), block(16, 16);
  pairwise_kernel<<<grid, block, 0, stream>>>(wsA, wsB, W2, b2, out);
}